// ComputeEdgeLoss_12506944766665
// MI455X (gfx1250) — compile-verified
//
#include <hip/hip_runtime.h>

typedef __attribute__((ext_vector_type(2))) float v2f;
typedef __attribute__((ext_vector_type(8))) float v8f;

#define NPTS  64
#define NGT   2048
#define NPAIR 2016            // 64*63/2
#define NINT  5
#define NQ    (NPAIR * NINT)  // 10080 query points per batch
#define QT    (NQ / 16)       // 630 query tiles per batch (exact)
#define WPB   8               // waves per block (256 threads, wave32)

__global__ __launch_bounds__(256)
void edge_min_dist_kernel(const float* __restrict__ recon,
                          const float* __restrict__ gt,
                          float* __restrict__ dmin)
{
    __shared__ float4 g4[NGT];  // 32KB: (x, y, z, ||g||^2) per gt point

    const int b   = blockIdx.y;
    const int tid = threadIdx.x;

    // Stage this batch's gt points into LDS, precomputing the squared norm
    // into the w slot (consumed by the WMMA's K=3 row against A's K3 == 1).
    const float* gb = gt + (size_t)b * NGT * 3;
    for (int pnt = tid; pnt < NGT; pnt += 256) {
        float x = gb[pnt * 3 + 0];
        float y = gb[pnt * 3 + 1];
        float z = gb[pnt * 3 + 2];
        g4[pnt] = make_float4(x, y, z, x * x + y * y + z * z);
    }
    __syncthreads();

    const int wave = tid >> 5;
    const int lane = tid & 31;
    const int tile = blockIdx.x * WPB + wave;
    if (tile >= QT) return;                 // wave-uniform: EXEC all-1s inside

    const bool hi  = (lane >= 16);
    const int  m   = lane & 15;             // query row (and gt column) slot
    const int  qidx = tile * 16 + m;
    const int  p   = qidx / NINT;
    const int  k   = qidx - p * NINT;

    // Invert lower-triangular pair index p -> (i, j), i > j (row-major tril).
    int i = (int)((1.0f + sqrtf(1.0f + 8.0f * (float)p)) * 0.5f);
    while (i * (i - 1) / 2 > p) --i;
    while ((i + 1) * i / 2 <= p) ++i;
    const int j = p - i * (i - 1) / 2;

    // Interpolated query point: start*f + end*(1-f), f = k/4.
    const float* rb = recon + (size_t)b * NPTS * 3;
    const float f  = 0.25f * (float)k;
    const float g1 = 1.0f - f;
    const float px = rb[i * 3 + 0] * f + rb[j * 3 + 0] * g1;
    const float py = rb[i * 3 + 1] * f + rb[j * 3 + 1] * g1;
    const float pz = rb[i * 3 + 2] * f + rb[j * 3 + 2] * g1;
    const float qn = px * px + py * py + pz * pz;

    // A matrix (16x4 f32): lanes 0-15 hold (K0,K1)=(-2x,-2y);
    // lanes 16-31 hold (K2,K3)=(-2z, 1).  With B's K3 row = ||g||^2 this makes
    // the WMMA compute D[m][n] = -2*q.g + ||g||^2 directly (C = 0 inline).
    v2f a;
    a.x = hi ? (-2.0f * pz) : (-2.0f * px);
    a.y = hi ? 1.0f         : (-2.0f * py);

    // Row norms for the C/D layout: VGPR r holds row m = r + 8*hi.
    const int roff = hi ? 8 : 0;
    float qr[8];
#pragma unroll
    for (int r = 0; r < 8; ++r) qr[r] = __shfl(qn, roff + r, 32);

    float mn[8];
#pragma unroll
    for (int r = 0; r < 8; ++r) mn[r] = 3.4e38f;

    const float2* g2   = (const float2*)g4;
    const int     boff = hi ? 1 : 0;        // (x,y) half vs (z,||g||^2) half

    for (int gtile = 0; gtile < NGT / 16; ++gtile) {
        // B matrix (4x16 f32): lanes 0-15 supply (K0,K1)=(gx,gy) of column m,
        // lanes 16-31 supply (K2,K3)=(gz,||g||^2).
        float2 bv = g2[(gtile * 16 + m) * 2 + boff];
        v2f bb; bb.x = bv.x; bb.y = bv.y;

        v8f c = {};  // inline zero accumulator
        v8f d = __builtin_amdgcn_wmma_f32_16x16x4_f32(
            false, a, false, bb, (short)0, c, false, false);

#pragma unroll
        for (int r = 0; r < 8; ++r) mn[r] = fminf(mn[r], d[r]);
    }

    // Min-reduce each row across its 16 columns-per-lane slice, then add ||q||^2.
#pragma unroll
    for (int r = 0; r < 8; ++r) {
        mn[r] = fminf(mn[r], __shfl_xor(mn[r], 1, 32));
        mn[r] = fminf(mn[r], __shfl_xor(mn[r], 2, 32));
        mn[r] = fminf(mn[r], __shfl_xor(mn[r], 4, 32));
        mn[r] = fminf(mn[r], __shfl_xor(mn[r], 8, 32));
        mn[r] += qr[r];
    }

    if (m == 0) {  // lane 0 writes rows 0-7, lane 16 writes rows 8-15
        float4* o4 = (float4*)(dmin + (size_t)b * NQ + tile * 16 + roff);
        o4[0] = make_float4(mn[0], mn[1], mn[2], mn[3]);
        o4[1] = make_float4(mn[4], mn[5], mn[6], mn[7]);
    }
}

__global__ __launch_bounds__(256)
void edge_mean5_kernel(const float* __restrict__ dmin,
                       float* __restrict__ out, int total)
{
    int idx = blockIdx.x * blockDim.x + threadIdx.x;
    if (idx >= total) return;
    int b = idx / NPAIR;
    int p = idx - b * NPAIR;
    const float* dp = dmin + (size_t)b * NQ + p * NINT;
    out[idx] = (dp[0] + dp[1] + dp[2] + dp[3] + dp[4]) * 0.2f;
}

extern "C" void kernel_launch(void* const* d_in, const int* in_sizes, int n_in,
                              void* d_out, int out_size, void* d_ws, size_t ws_size,
                              hipStream_t stream)
{
    const float* recon = (const float*)d_in[0];   // [B, 64, 3] f32
    const float* gt    = (const float*)d_in[1];   // [B, 2048, 3] f32
    float*       out   = (float*)d_out;           // [B, 2016] f32
    float*       dmin  = (float*)d_ws;            // [B, 10080] f32 scratch

    const int B = in_sizes[0] / (NPTS * 3);       // 4

    dim3 grid((QT + WPB - 1) / WPB, B, 1);        // 79 x 4 blocks, 256 thr
    edge_min_dist_kernel<<<grid, 256, 0, stream>>>(recon, gt, dmin);

    const int total = B * NPAIR;
    edge_mean5_kernel<<<(total + 255) / 256, 256, 0, stream>>>(dmin, out, total);
}